// LorentzMultiHeadAttention_3384434229837
// MI455X (gfx1250) — compile-verified
//
#include <hip/hip_runtime.h>
#include <hip/hip_bf16.h>
#include <math.h>

// ---------------------------------------------------------------------------
// Lorentz multi-head attention, fp32, CDNA5 (gfx1250) WMMA f32 16x16x4 path.
// Memory-bound by mandatory attn_w materialization (~2.1 GB HBM traffic),
// so fp32 matrix pipe is the right precision/perf point on MI455X.
// ---------------------------------------------------------------------------

typedef __attribute__((ext_vector_type(2))) float v2f;
typedef __attribute__((ext_vector_type(8))) float v8f;

#define D_MODEL   1024
#define N_HEADS   16
#define HEAD_DIM  64
#define SEQ_LEN   2048
#define BATCH     2
#define M_TOK     (BATCH * SEQ_LEN)            // 4096
#define BH        (BATCH * N_HEADS)            // 32
#define ALPHA_F   0.25f
#define INV_SCALE 0.125f                       // 1/sqrt(64)

#define LDA 36   // K-contiguous LDS tile pad (rowK fragments, conflict-free)
#define LDB 80   // K-major V tile pad (colK fragments, conflict-free)

__device__ __forceinline__ v8f wmma_f32(v2f a, v2f b, v8f c) {
  // D = A(16x4,f32) * B(4x16,f32) + C(16x16,f32)
  return __builtin_amdgcn_wmma_f32_16x16x4_f32(
      /*neg_a=*/false, a, /*neg_b=*/false, b,
      /*c_mod=*/(short)0, c, /*reuse_a=*/false, /*reuse_b=*/false);
}

// rowK fragment: element [row r][k] lives at p[r*ld + k].
// Lane l: r = l&15, k = 2*(l>>4) + v  (v = vgpr 0/1)
__device__ __forceinline__ v2f frag_rowK(const float* p, int ld, int lane) {
  const float* q = p + (lane & 15) * ld + ((lane >> 4) << 1);
  v2f r; r.x = q[0]; r.y = q[1];
  return r;
}

// colK fragment from K-major tile: element B[k][n] at p[k*ld + n].
__device__ __forceinline__ v2f frag_colK(const float* p, int ld, int lane) {
  int n  = lane & 15;
  int kh = (lane >> 4) << 1;
  v2f r; r.x = p[kh * ld + n]; r.y = p[(kh + 1) * ld + n];
  return r;
}

// ---------------------------------------------------------------------------
// Kernel 1: projection  out(b,h,l,d) = sum_k x[(b,l),k] * W[n,k], n = h*64+d
// Grid: (M_TOK/64, D_MODEL/64), block 256 (8 waves; wave = 32x16 tile x2 M)
// ---------------------------------------------------------------------------
__global__ void lmha_proj_kernel(const float* __restrict__ x,
                                 const float* __restrict__ W,
                                 float* __restrict__ out) {
  __shared__ float As[64 * LDA];
  __shared__ float Bs[64 * LDA];
  const int m0 = blockIdx.x * 64;
  const int n0 = blockIdx.y * 64;
  const int tid = threadIdx.x, lane = tid & 31, wave = tid >> 5;
  const int wm = wave >> 2, wn = wave & 3;

  v8f acc0 = {}; v8f acc1 = {};

  for (int k0 = 0; k0 < D_MODEL; k0 += 32) {
    __syncthreads();
    for (int s = tid; s < 512; s += 256) {
      int r = s >> 3, c = (s & 7) << 2;
      float4 va = *(const float4*)(x + (size_t)(m0 + r) * D_MODEL + k0 + c);
      float* da = As + r * LDA + c;
      da[0] = va.x; da[1] = va.y; da[2] = va.z; da[3] = va.w;
      float4 vb = *(const float4*)(W + (size_t)(n0 + r) * D_MODEL + k0 + c);
      float* db = Bs + r * LDA + c;
      db[0] = vb.x; db[1] = vb.y; db[2] = vb.z; db[3] = vb.w;
    }
    __syncthreads();
    const float* ap0 = As + (wm * 32) * LDA;
    const float* ap1 = As + (wm * 32 + 16) * LDA;
    const float* bp  = Bs + (wn * 16) * LDA;
#pragma unroll
    for (int kk = 0; kk < 32; kk += 4) {
      v2f bf  = frag_rowK(bp  + kk, LDA, lane);   // B[k][n] = W[n][k]
      v2f af0 = frag_rowK(ap0 + kk, LDA, lane);
      v2f af1 = frag_rowK(ap1 + kk, LDA, lane);
      acc0 = wmma_f32(af0, bf, acc0);
      acc1 = wmma_f32(af1, bf, acc1);
    }
  }

  // epilogue: split-head layout out[((b*16+h)*2048 + l)*64 + d]
  const int n = n0 + wn * 16 + (lane & 15);
  const int h = n >> 6, d = n & 63;
  const int mb = m0 + wm * 32 + (lane >> 4) * 8;
#pragma unroll
  for (int v = 0; v < 8; ++v) {
    int m = mb + v;
    int b = m >> 11, l = m & 2047;
    out[((size_t)(b * N_HEADS + h) * SEQ_LEN + l) * HEAD_DIM + d] = acc0[v];
  }
#pragma unroll
  for (int v = 0; v < 8; ++v) {
    int m = mb + 16 + v;
    int b = m >> 11, l = m & 2047;
    out[((size_t)(b * N_HEADS + h) * SEQ_LEN + l) * HEAD_DIM + d] = acc1[v];
  }
}

// ---------------------------------------------------------------------------
// Kernel 2: fold Lorentz time term + 1/scale into Q (in place).
// Qeff[d] = Q[d] * (1 - 2*alpha*sf*mask[d]) / sqrt(d_h)
// Block 256 = 4 rows of 64; grid BH*SEQ_LEN/4.
// ---------------------------------------------------------------------------
__global__ void lmha_qeff_kernel(float* __restrict__ Q,
                                 const unsigned char* __restrict__ tmask) {
  __shared__ float s1[256];
  __shared__ float s2[256];
  const int tid = threadIdx.x;
  const int grp = tid >> 6;
  const int d   = tid & 63;
  const long row = (long)blockIdx.x * 4 + grp;      // over BH*SEQ_LEN
  const int h = (int)((row >> 11) & (N_HEADS - 1));
  const float mk = tmask[h * HEAD_DIM + d] ? 1.0f : 0.0f;
  const float q = Q[row * HEAD_DIM + d];
  s1[tid] = q * q;
  s2[tid] = q * q * mk;
  __syncthreads();
#pragma unroll
  for (int off = 32; off > 0; off >>= 1) {
    if (d < off) { s1[tid] += s1[tid + off]; s2[tid] += s2[tid + off]; }
    __syncthreads();
  }
  const float qn  = sqrtf(s1[grp * 64]);
  const float qtn = sqrtf(s2[grp * 64]);
  const float sf  = (qtn > 1e-6f) ? (qn / fmaxf(qtn, 1e-8f)) : 0.0f;
  Q[row * HEAD_DIM + d] = q * (1.0f - 2.0f * ALPHA_F * sf * mk) * INV_SCALE;
}

// ---------------------------------------------------------------------------
// Kernel 3: scores[q,k] = Qeff[q,:].K[k,:] + mask[b,k]   (per b,h)
// M=N=2048, K=64. Both operands K-contiguous -> both rowK fragments.
// Grid: (32, 32, BH).
// ---------------------------------------------------------------------------
__global__ void lmha_scores_kernel(const float* __restrict__ Qe,
                                   const float* __restrict__ Kb,
                                   const float* __restrict__ mask,
                                   float* __restrict__ attn) {
  __shared__ float As[64 * LDA];
  __shared__ float Bs[64 * LDA];
  const int bh = blockIdx.z, b = bh >> 4;
  const float* Abase = Qe + (size_t)bh * SEQ_LEN * HEAD_DIM;
  const float* Bbase = Kb + (size_t)bh * SEQ_LEN * HEAD_DIM;
  const int m0 = blockIdx.x * 64;
  const int n0 = blockIdx.y * 64;
  const int tid = threadIdx.x, lane = tid & 31, wave = tid >> 5;
  const int wm = wave >> 2, wn = wave & 3;

  v8f acc0 = {}; v8f acc1 = {};

  for (int k0 = 0; k0 < HEAD_DIM; k0 += 32) {
    __syncthreads();
    for (int s = tid; s < 512; s += 256) {
      int r = s >> 3, c = (s & 7) << 2;
      float4 va = *(const float4*)(Abase + (size_t)(m0 + r) * HEAD_DIM + k0 + c);
      float* da = As + r * LDA + c;
      da[0] = va.x; da[1] = va.y; da[2] = va.z; da[3] = va.w;
      float4 vb = *(const float4*)(Bbase + (size_t)(n0 + r) * HEAD_DIM + k0 + c);
      float* db = Bs + r * LDA + c;
      db[0] = vb.x; db[1] = vb.y; db[2] = vb.z; db[3] = vb.w;
    }
    __syncthreads();
    const float* ap0 = As + (wm * 32) * LDA;
    const float* ap1 = As + (wm * 32 + 16) * LDA;
    const float* bp  = Bs + (wn * 16) * LDA;
#pragma unroll
    for (int kk = 0; kk < 32; kk += 4) {
      v2f bf  = frag_rowK(bp  + kk, LDA, lane);
      v2f af0 = frag_rowK(ap0 + kk, LDA, lane);
      v2f af1 = frag_rowK(ap1 + kk, LDA, lane);
      acc0 = wmma_f32(af0, bf, acc0);
      acc1 = wmma_f32(af1, bf, acc1);
    }
  }

  const int n = n0 + wn * 16 + (lane & 15);
  const float mval = mask[b * SEQ_LEN + n];
  const int mb = m0 + wm * 32 + (lane >> 4) * 8;
  float* obase = attn + (size_t)bh * SEQ_LEN * SEQ_LEN;
#pragma unroll
  for (int v = 0; v < 8; ++v)
    obase[(size_t)(mb + v) * SEQ_LEN + n] = acc0[v] + mval;
#pragma unroll
  for (int v = 0; v < 8; ++v)
    obase[(size_t)(mb + 16 + v) * SEQ_LEN + n] = acc1[v] + mval;
}

// ---------------------------------------------------------------------------
// Kernel 4: in-place softmax over rows of 2048. One 256-block per row.
// ---------------------------------------------------------------------------
__global__ void lmha_softmax_kernel(float* __restrict__ attn) {
  __shared__ float red[256];
  const int tid = threadIdx.x;
  float* p = attn + (size_t)blockIdx.x * SEQ_LEN;
  float vals[8];
  float mx = -3.4e38f;
#pragma unroll
  for (int i = 0; i < 8; ++i) {
    vals[i] = p[tid + i * 256];
    mx = fmaxf(mx, vals[i]);
  }
  red[tid] = mx; __syncthreads();
#pragma unroll
  for (int off = 128; off > 0; off >>= 1) {
    if (tid < off) red[tid] = fmaxf(red[tid], red[tid + off]);
    __syncthreads();
  }
  mx = red[0]; __syncthreads();
  float sum = 0.0f;
#pragma unroll
  for (int i = 0; i < 8; ++i) { vals[i] = expf(vals[i] - mx); sum += vals[i]; }
  red[tid] = sum; __syncthreads();
#pragma unroll
  for (int off = 128; off > 0; off >>= 1) {
    if (tid < off) red[tid] += red[tid + off];
    __syncthreads();
  }
  const float inv = 1.0f / red[0];
#pragma unroll
  for (int i = 0; i < 8; ++i) p[tid + i * 256] = vals[i] * inv;
}

// ---------------------------------------------------------------------------
// Kernel 5: O[q,d] = sum_k attn[q,k] * V[k,d]   (per b,h)
// M=2048, N=64, K=2048. A rowK frags, B colK frags (V is K-major).
// Grid: (32, 1, BH).
// ---------------------------------------------------------------------------
__global__ void lmha_attnv_kernel(const float* __restrict__ attn,
                                  const float* __restrict__ Vb,
                                  float* __restrict__ O) {
  __shared__ float As[64 * LDA];
  __shared__ float Bs[32 * LDB];
  const int bh = blockIdx.z;
  const float* Abase = attn + (size_t)bh * SEQ_LEN * SEQ_LEN;
  const float* Bbase = Vb + (size_t)bh * SEQ_LEN * HEAD_DIM;
  const int m0 = blockIdx.x * 64;
  const int tid = threadIdx.x, lane = tid & 31, wave = tid >> 5;
  const int wm = wave >> 2, wn = wave & 3;

  v8f acc0 = {}; v8f acc1 = {};

  for (int k0 = 0; k0 < SEQ_LEN; k0 += 32) {
    __syncthreads();
    for (int s = tid; s < 512; s += 256) {
      int r = s >> 3, c = (s & 7) << 2;
      float4 va = *(const float4*)(Abase + (size_t)(m0 + r) * SEQ_LEN + k0 + c);
      float* da = As + r * LDA + c;
      da[0] = va.x; da[1] = va.y; da[2] = va.z; da[3] = va.w;
    }
    for (int s = tid; s < 512; s += 256) {        // 32 rows x 16 float4
      int r = s >> 4, c = (s & 15) << 2;
      float4 vb = *(const float4*)(Bbase + (size_t)(k0 + r) * HEAD_DIM + c);
      float* db = Bs + r * LDB + c;
      db[0] = vb.x; db[1] = vb.y; db[2] = vb.z; db[3] = vb.w;
    }
    __syncthreads();
    const float* ap0 = As + (wm * 32) * LDA;
    const float* ap1 = As + (wm * 32 + 16) * LDA;
    const float* bp  = Bs + wn * 16;              // n offset within 64
#pragma unroll
    for (int kk = 0; kk < 32; kk += 4) {
      v2f bf  = frag_colK(bp + kk * LDB, LDB, lane);
      v2f af0 = frag_rowK(ap0 + kk, LDA, lane);
      v2f af1 = frag_rowK(ap1 + kk, LDA, lane);
      acc0 = wmma_f32(af0, bf, acc0);
      acc1 = wmma_f32(af1, bf, acc1);
    }
  }

  const int d = wn * 16 + (lane & 15);
  const int mb = m0 + wm * 32 + (lane >> 4) * 8;
  float* obase = O + (size_t)bh * SEQ_LEN * HEAD_DIM;
#pragma unroll
  for (int v = 0; v < 8; ++v)
    obase[(size_t)(mb + v) * HEAD_DIM + d] = acc0[v];
#pragma unroll
  for (int v = 0; v < 8; ++v)
    obase[(size_t)(mb + 16 + v) * HEAD_DIM + d] = acc1[v];
}

// ---------------------------------------------------------------------------
// Kernel 6: final[m,n] = sum_k Oflat[m,k] * Wo[n,k], with
// Oflat[(b,l), h*64+d] = O[b,h,l,d] folded into the stage-in addressing.
// Grid: (64, 16).
// ---------------------------------------------------------------------------
__global__ void lmha_outproj_kernel(const float* __restrict__ O,
                                    const float* __restrict__ Wo,
                                    float* __restrict__ out) {
  __shared__ float As[64 * LDA];
  __shared__ float Bs[64 * LDA];
  const int m0 = blockIdx.x * 64;
  const int n0 = blockIdx.y * 64;
  const int tid = threadIdx.x, lane = tid & 31, wave = tid >> 5;
  const int wm = wave >> 2, wn = wave & 3;
  const int bb = m0 >> 11, lbase = m0 & 2047;

  v8f acc0 = {}; v8f acc1 = {};

  for (int k0 = 0; k0 < D_MODEL; k0 += 32) {
    const int h = k0 >> 6, dbase = k0 & 63;       // 32-chunk stays in one head
    __syncthreads();
    for (int s = tid; s < 512; s += 256) {
      int r = s >> 3, c = (s & 7) << 2;
      float4 va = *(const float4*)(O +
          ((size_t)(bb * N_HEADS + h) * SEQ_LEN + lbase + r) * HEAD_DIM + dbase + c);
      float* da = As + r * LDA + c;
      da[0] = va.x; da[1] = va.y; da[2] = va.z; da[3] = va.w;
      float4 vb = *(const float4*)(Wo + (size_t)(n0 + r) * D_MODEL + k0 + c);
      float* db = Bs + r * LDA + c;
      db[0] = vb.x; db[1] = vb.y; db[2] = vb.z; db[3] = vb.w;
    }
    __syncthreads();
    const float* ap0 = As + (wm * 32) * LDA;
    const float* ap1 = As + (wm * 32 + 16) * LDA;
    const float* bp  = Bs + (wn * 16) * LDA;
#pragma unroll
    for (int kk = 0; kk < 32; kk += 4) {
      v2f bf  = frag_rowK(bp  + kk, LDA, lane);
      v2f af0 = frag_rowK(ap0 + kk, LDA, lane);
      v2f af1 = frag_rowK(ap1 + kk, LDA, lane);
      acc0 = wmma_f32(af0, bf, acc0);
      acc1 = wmma_f32(af1, bf, acc1);
    }
  }

  const int n = n0 + wn * 16 + (lane & 15);
  const int mb = m0 + wm * 32 + (lane >> 4) * 8;
#pragma unroll
  for (int v = 0; v < 8; ++v)
    out[(size_t)(mb + v) * D_MODEL + n] = acc0[v];
#pragma unroll
  for (int v = 0; v < 8; ++v)
    out[(size_t)(mb + 16 + v) * D_MODEL + n] = acc1[v];
}

// ---------------------------------------------------------------------------
extern "C" void kernel_launch(void* const* d_in, const int* in_sizes, int n_in,
                              void* d_out, int out_size, void* d_ws, size_t ws_size,
                              hipStream_t stream) {
  const float* x     = (const float*)d_in[0];
  const float* amask = (const float*)d_in[1];
  const unsigned char* tmask = (const unsigned char*)d_in[2];
  const float* Wq = (const float*)d_in[3];
  const float* Wk = (const float*)d_in[4];
  const float* Wv = (const float*)d_in[5];
  const float* Wo = (const float*)d_in[6];

  const size_t QSZ = (size_t)BH * SEQ_LEN * HEAD_DIM;   // 4,194,304 floats
  float* Q = (float*)d_ws;
  float* K = Q + QSZ;
  float* V = K + QSZ;
  float* O = Q;                                         // reuse Q after scores

  float* out_final = (float*)d_out;                     // (B, L, D_MODEL)
  float* attn      = out_final + (size_t)M_TOK * D_MODEL;  // (B, H, L, L)

  dim3 blk(256);

  // 1) Q/K/V projections (split-head layout)
  dim3 gproj(M_TOK / 64, D_MODEL / 64);
  lmha_proj_kernel<<<gproj, blk, 0, stream>>>(x, Wq, Q);
  lmha_proj_kernel<<<gproj, blk, 0, stream>>>(x, Wk, K);
  lmha_proj_kernel<<<gproj, blk, 0, stream>>>(x, Wv, V);

  // 2) fold Lorentz time term + 1/scale into Q
  lmha_qeff_kernel<<<dim3(BH * SEQ_LEN / 4), blk, 0, stream>>>(Q, tmask);

  // 3) scores -> attn_w slice of d_out
  dim3 gsc(SEQ_LEN / 64, SEQ_LEN / 64, BH);
  lmha_scores_kernel<<<gsc, blk, 0, stream>>>(Q, K, amask, attn);

  // 4) softmax in place
  lmha_softmax_kernel<<<dim3(BH * SEQ_LEN), blk, 0, stream>>>(attn);

  // 5) attn @ V
  dim3 gav(SEQ_LEN / 64, 1, BH);
  lmha_attnv_kernel<<<gav, blk, 0, stream>>>(attn, V, O);

  // 6) output projection
  dim3 gout(M_TOK / 64, D_MODEL / 64);
  lmha_outproj_kernel<<<gout, blk, 0, stream>>>(O, Wo, out_final);
}